// CustomPoisson_12292196401945
// MI455X (gfx1250) — compile-verified
//
#include <hip/hip_runtime.h>
#include <hip/hip_bf16.h>
#include <stdint.h>

// Poisson spike encoding: out[k*N + i] = (cummax_{j<=k}(j + t_j) > k) ? 1.0f : 0.0f
// Streaming form per pixel: rem = max(rem - 1, t_k); spike = rem > 0.
// Output (524 MB f32) is streamed to HBM via the CDNA5 Tensor Data Mover:
// each wave double-buffers [ROWS x 32] f32 tiles in LDS and issues
// tensor_store_from_lds (TENSORcnt-tracked) while computing the next tile.

#define T_WINDOW 500
#define ROWS     20                  // rows per TDM tile (500 % 20 == 0)
#define CHUNKS   (T_WINDOW / ROWS)   // 25 TDM stores per wave
#define BLOCK    256
#define WAVES    (BLOCK / 32)

typedef __attribute__((ext_vector_type(4))) unsigned int v4u;
typedef __attribute__((ext_vector_type(4))) int          v4i;
typedef __attribute__((ext_vector_type(8))) int          v8i;

// Low 32 bits of a generic pointer to LDS == LDS byte offset (aperture mapping).
__device__ __forceinline__ uint32_t lds_off(const void* p) {
    return (uint32_t)(uintptr_t)p;
}

__device__ __forceinline__ float rng_uniform(uint32_t& s) {
    s ^= s << 13; s ^= s >> 17; s ^= s << 5;          // xorshift32, ~4 VALU ops
    return (float)(s >> 8) * 0x1.0p-24f;              // [0,1)
}

// Inverse-CDF Poisson draw: exactly one uniform; expected extra iterations = lambda (< 1).
__device__ __forceinline__ int poisson_icdf(float lam, float exp_neg_lam, uint32_t& s) {
    float u   = rng_uniform(s);
    float cdf = exp_neg_lam;
    float pmf = exp_neg_lam;
    int   t   = 0;
#pragma unroll 1
    while (u > cdf && t < 12) {                       // P(t>12 | lam<1) ~ 1e-10
        t += 1;
        pmf *= lam * __frcp_rn((float)t);
        cdf += pmf;
    }
    return t;
}

// Issue one 2D TDM store: LDS tile [tile_y rows x tile_x elems] (contiguous, 4B elems)
// -> global rows of tile_x*4 bytes with pitch n_elems*4 bytes.
__device__ __forceinline__ void tdm_store_tile_2d(uint32_t lds_addr, uint64_t gaddr,
                                                  uint32_t n_elems, uint32_t t_rows,
                                                  uint32_t tile_x, uint32_t tile_y) {
#if defined(__AMDGCN__)
    // ---- D# group 0 (128b): count=1 | lds_addr | global_addr[56:0] | type=2 ----
    v4u g0;
    g0.x = 1u;                                                   // count=1, user mode
    g0.y = lds_addr;                                             // LDS byte address
    g0.z = (uint32_t)gaddr;                                      // global addr [31:0]
    g0.w = ((uint32_t)(gaddr >> 32) & 0x01FFFFFFu) | (2u << 30); // [56:32] | type=2
    // ---- D# group 1 (256b) ----
    v8i g1;
    g1[0] = (int)(2u << 16);                                     // wg_mask=0, data_size=4B
    g1[1] = (int)((n_elems & 0xFFFFu) << 16);                    // tensor_dim0[15:0]
    g1[2] = (int)((n_elems >> 16) | ((t_rows & 0xFFFFu) << 16)); // dim0[31:16] | dim1[15:0]
    g1[3] = (int)((t_rows >> 16) | ((tile_x & 0xFFFFu) << 16));  // dim1[31:16] | tile_dim0
    g1[4] = (int)(tile_y & 0xFFFFu);                             // tile_dim1 | tile_dim2=0
    g1[5] = (int)n_elems;                                        // tensor_dim0_stride[31:0]
    g1[6] = 0;                                                   // stride[47:32] | dim1_stride lo
    g1[7] = 0;
    v4i gz = {0, 0, 0, 0};                                       // groups 2/3 unused (2D)
#if defined(__has_builtin)
#if __has_builtin(__builtin_amdgcn_tensor_store_from_lds)
#if __clang_major__ >= 23
    v8i gz8 = {0, 0, 0, 0, 0, 0, 0, 0};
    __builtin_amdgcn_tensor_store_from_lds(g0, g1, gz, gz, gz8, 0);
#else
    __builtin_amdgcn_tensor_store_from_lds(g0, g1, gz, gz, 0);
#endif
#else
    asm volatile("tensor_store_from_lds %0, %1, %2, %3"
                 :: "s"(g0), "s"(g1), "s"(gz), "s"(gz) : "memory");
#endif
#else
    asm volatile("tensor_store_from_lds %0, %1, %2, %3"
                 :: "s"(g0), "s"(g1), "s"(gz), "s"(gz) : "memory");
#endif
#endif // __AMDGCN__
}

__global__ __launch_bounds__(BLOCK)
void poisson_encode_kernel(const float* __restrict__ img, float* __restrict__ out,
                           uint32_t N) {
    // Per-wave private double-buffered tiles: [wave][buf][row][lane]
    __shared__ float tile[WAVES][2][ROWS][32];

    const uint32_t tid  = threadIdx.x;
    const uint32_t lane = tid & 31u;
    const uint32_t w    = (uint32_t)__builtin_amdgcn_readfirstlane((int)(tid >> 5));
    const uint32_t pix  = blockIdx.x * BLOCK + tid;

    const float lam = img[pix];
    const float enl = __expf(-lam);

    // Per-pixel deterministic seed (murmur-ish mix), never zero.
    uint32_t s = pix * 2654435761u + 0x9E3779B9u;
    s ^= s >> 16; s *= 0x85EBCA6Bu; s ^= s >> 13; s |= 1u;

    int rem = 0;  // rem_k = run_max_k - k; spike iff rem > 0

    // Global byte address of this wave's 32-pixel column block, row 0.
    uint64_t gaddr = (uint64_t)(uintptr_t)out +
                     4ull * (uint64_t)(blockIdx.x * BLOCK + (w << 5));
    const uint64_t gstep = 4ull * (uint64_t)ROWS * (uint64_t)N;

    uint32_t buf = 0;
    for (int c = 0; c < CHUNKS; ++c) {
        // Buffer reuse gate: allow the other buffer's TDM store to remain in flight.
#if defined(__AMDGCN__)
        asm volatile("s_wait_tensorcnt 0x1" ::: "memory");
#endif
#pragma unroll 4
        for (int r = 0; r < ROWS; ++r) {
            const int t = poisson_icdf(lam, enl, s);
            const int d = rem - 1;
            rem = (d > t) ? d : t;
            tile[w][buf][r][lane] = (rem > 0) ? 1.0f : 0.0f;   // ds_store_b32, bank-clean
        }
#if defined(__AMDGCN__)
        // Make this wave's LDS writes visible before the TDM engine reads them.
        asm volatile("s_wait_dscnt 0x0" ::: "memory");
#endif
        tdm_store_tile_2d(lds_off(&tile[w][buf][0][0]), gaddr,
                          N, (uint32_t)T_WINDOW, 32u, (uint32_t)ROWS);
        gaddr += gstep;
        buf ^= 1u;
    }
#if defined(__AMDGCN__)
    asm volatile("s_wait_tensorcnt 0x0" ::: "memory");
#endif
}

extern "C" void kernel_launch(void* const* d_in, const int* in_sizes, int n_in,
                              void* d_out, int out_size, void* d_ws, size_t ws_size,
                              hipStream_t stream) {
    (void)n_in; (void)out_size; (void)d_ws; (void)ws_size;
    const float* img = (const float*)d_in[0];
    float*       out = (float*)d_out;
    const uint32_t N = (uint32_t)in_sizes[0];        // 262144, divisible by 256
    const uint32_t blocks = N / BLOCK;               // 1024 workgroups of 8 waves
    poisson_encode_kernel<<<blocks, BLOCK, 0, stream>>>(img, out, N);
}